// DevNet_63093069578584
// MI455X (gfx1250) — compile-verified
//
#include <hip/hip_runtime.h>

typedef __attribute__((ext_vector_type(2))) float v2f;
typedef __attribute__((ext_vector_type(8))) float v8f;

#define IN_DIM 128   // input feature dim == H*D output dim
#define D_DIM  64    // per-head dim
#define CAP_E  96    // max edges used per direction (Poisson(16): P(>96) ~ 1e-60)
#define MAXR   112   // (CAP_E+1) rounded up to multiple of 16

// ---------------- zero the two edge counters ----------------
__global__ void zero_cnt_kernel(int* cnt) {
    if (threadIdx.x < 2) cnt[threadIdx.x] = 0;
}

// ---------------- scan: collect edges touching node op ----------------
__global__ __launch_bounds__(256)
void scan_kernel(const int* __restrict__ src, const int* __restrict__ dst,
                 const int* __restrict__ op_ptr, int E,
                 int* __restrict__ cnt, int* __restrict__ listF,
                 int* __restrict__ listB, int cap) {
    const int op = op_ptr[0];
    for (int e = blockIdx.x * blockDim.x + threadIdx.x; e < E;
         e += gridDim.x * blockDim.x) {
        if (dst[e] == op) { int i = atomicAdd(&cnt[0], 1); if (i < cap) listF[i] = e; }
        if (src[e] == op) { int i = atomicAdd(&cnt[1], 1); if (i < cap) listB[i] = e; }
    }
}

// ---------------- single-WG: WMMA GEMM on gathered rows + attention ----------------
__global__ __launch_bounds__(256)
void gat_kernel(const float* __restrict__ feat,
                const float* __restrict__ W_f, const float* __restrict__ aL_f,
                const float* __restrict__ aR_f, const float* __restrict__ b_f,
                const float* __restrict__ W_b, const float* __restrict__ aL_b,
                const float* __restrict__ aR_b, const float* __restrict__ b_b,
                const int* __restrict__ src, const int* __restrict__ dst,
                const int* __restrict__ op_ptr, const int* __restrict__ parallel,
                const int* __restrict__ cnt, const int* __restrict__ listF,
                const int* __restrict__ listB, int cap_ws,
                float* __restrict__ out)
{
    __shared__ float Hbuf[MAXR][IN_DIM];   // 56 KB: h rows for [op, neighbors...]
    __shared__ int   s_list[CAP_E];
    __shared__ int   s_nodes[MAXR];
    __shared__ float s_el[MAXR][2];
    __shared__ float s_ev[CAP_E][2];
    __shared__ float s_er0[2];
    __shared__ int   s_M;

    const int  tid  = threadIdx.x;
    const int  lane = tid & 31;
    const int  wave = tid >> 5;          // 0..7, one 16-col tile each
    const int  m16  = lane & 15;
    const bool hilf = (lane >= 16);
    const int  op   = op_ptr[0];

    for (int dir = 0; dir < 2; ++dir) {
        const float* Wd = dir ? W_b  : W_f;
        const float* aL = dir ? aL_b : aL_f;
        const float* aR = dir ? aR_b : aR_f;
        const float* bs = dir ? b_b  : b_f;
        const int*   ls = dir ? listB : listF;

        if (tid == 0) {
            int c = cnt[dir];
            if (c > cap_ws) c = cap_ws;
            if (c > CAP_E)  c = CAP_E;
            s_M = c;
        }
        __syncthreads();
        const int M = s_M;
        if (tid < M) s_list[tid] = ls[tid];
        __syncthreads();
        if (tid == 0) {            // sort edge ids -> deterministic reductions
            for (int i = 1; i < M; ++i) {
                int v = s_list[i]; int j = i - 1;
                while (j >= 0 && s_list[j] > v) { s_list[j + 1] = s_list[j]; --j; }
                s_list[j + 1] = v;
            }
        }
        __syncthreads();
        for (int i = tid; i < MAXR; i += blockDim.x) {
            int node = op;                       // row 0 = op; pad rows = op
            if (i >= 1 && i <= M) { int e = s_list[i - 1]; node = dir ? dst[e] : src[e]; }
            s_nodes[i] = node;
        }
        __syncthreads();

        // ---- H = featGathered @ Wd via V_WMMA_F32_16X16X4_F32 ----
        const int R  = M + 1;
        const int RT = (R + 15) >> 4;
        const int n  = wave * 16 + m16;          // output column 0..127
        const int kb = hilf ? 2 : 0;             // K offset for high half-wave
        for (int rt = 0; rt < RT; ++rt) {
            const int node = s_nodes[rt * 16 + m16];
            const float* arow = feat + (size_t)node * IN_DIM + kb;
            v8f acc = {0.f, 0.f, 0.f, 0.f, 0.f, 0.f, 0.f, 0.f};
            #pragma unroll 8
            for (int k0 = 0; k0 < IN_DIM; k0 += 4) {
                // A 16x4 f32: lanes 0-15 K={k0,k0+1}, lanes 16-31 K={k0+2,k0+3}
                v2f a; a.x = arow[k0]; a.y = arow[k0 + 1];
                // B 4x16 f32: lanes 0-15 K={k0,k0+1}, lanes 16-31 K={k0+2,k0+3}, col n
                v2f b; b.x = Wd[(k0 + kb) * IN_DIM + n];
                       b.y = Wd[(k0 + kb + 1) * IN_DIM + n];
                acc = __builtin_amdgcn_wmma_f32_16x16x4_f32(
                          false, a, false, b, (short)0, acc, false, false);
            }
            const int rbase = rt * 16 + (hilf ? 8 : 0);
            #pragma unroll
            for (int i = 0; i < 8; ++i) Hbuf[rbase + i][n] = acc[i];
        }
        __syncthreads();

        // ---- per-row attention dots ----
        if (tid < R) {
            float e0 = 0.f, e1 = 0.f;
            for (int d = 0; d < D_DIM; ++d) {
                e0 += Hbuf[tid][d]         * aL[d];
                e1 += Hbuf[tid][D_DIM + d] * aL[D_DIM + d];
            }
            s_el[tid][0] = e0; s_el[tid][1] = e1;
            if (tid == 0) {
                float r0 = 0.f, r1 = 0.f;
                for (int d = 0; d < D_DIM; ++d) {
                    r0 += Hbuf[0][d]         * aR[d];
                    r1 += Hbuf[0][D_DIM + d] * aR[D_DIM + d];
                }
                s_er0[0] = r0; s_er0[1] = r1;
            }
        }
        __syncthreads();

        // ---- edge softmax over in-edges of op (per head) ----
        if (tid == 0) {
            for (int h = 0; h < 2; ++h) {
                float mx = -__builtin_inff();
                for (int i = 0; i < M; ++i) {
                    float v = s_el[i + 1][h] + s_er0[h];
                    v = (v >= 0.f) ? v : 0.2f * v;      // leaky_relu 0.2
                    s_ev[i][h] = v;
                    mx = fmaxf(mx, v);
                }
                float den = 0.f;
                for (int i = 0; i < M; ++i) {
                    float x = expf(s_ev[i][h] - mx);
                    s_ev[i][h] = x; den += x;
                }
                for (int i = 0; i < M; ++i) s_ev[i][h] /= den;
            }
        }
        __syncthreads();

        // ---- weighted sum + bias, mean over heads ----
        if (tid < D_DIM) {
            float a0 = 0.f, a1 = 0.f;
            for (int i = 0; i < M; ++i) {
                a0 += s_ev[i][0] * Hbuf[i + 1][tid];
                a1 += s_ev[i][1] * Hbuf[i + 1][D_DIM + tid];
            }
            out[dir * D_DIM + tid] = 0.5f * ((a0 + bs[tid]) + (a1 + bs[D_DIM + tid]));
        }
        __syncthreads();   // protect Hbuf/s_* before next direction
    }

    // ---- tails: feat[op] and sum(feat[parallel]) ----
    if (tid < IN_DIM) {
        out[2 * D_DIM + tid] = feat[(size_t)op * IN_DIM + tid];
    } else {
        const int j = tid - IN_DIM;
        float s = 0.f;
        for (int p = 0; p < 64; ++p) s += feat[(size_t)parallel[p] * IN_DIM + j];
        out[2 * D_DIM + IN_DIM + j] = s;
    }
}

extern "C" void kernel_launch(void* const* d_in, const int* in_sizes, int n_in,
                              void* d_out, int out_size, void* d_ws, size_t ws_size,
                              hipStream_t stream) {
    (void)n_in; (void)out_size;
    const float* feat = (const float*)d_in[0];
    const float* W_f  = (const float*)d_in[1];
    const float* aL_f = (const float*)d_in[2];
    const float* aR_f = (const float*)d_in[3];
    const float* b_f  = (const float*)d_in[4];
    const float* W_b  = (const float*)d_in[5];
    const float* aL_b = (const float*)d_in[6];
    const float* aR_b = (const float*)d_in[7];
    const float* b_b  = (const float*)d_in[8];
    const int*   src  = (const int*)d_in[9];
    const int*   dst  = (const int*)d_in[10];
    const int*   op   = (const int*)d_in[11];
    const int*   par  = (const int*)d_in[12];
    const int    E    = in_sizes[9];

    int cap_ws = 1024;
    {
        size_t need = (size_t)(8 + 2 * cap_ws) * sizeof(int);
        if (ws_size < need) {
            long avail = ((long)(ws_size / sizeof(int)) - 8) / 2;
            cap_ws = (avail > 0) ? (avail < 1024 ? (int)avail : 1024) : 0;
        }
    }
    int* cnt   = (int*)d_ws;
    int* listF = cnt + 8;
    int* listB = cnt + 8 + cap_ws;

    zero_cnt_kernel<<<1, 32, 0, stream>>>(cnt);
    int blocks = (E + 255) / 256;
    if (blocks > 1024) blocks = 1024;
    if (blocks < 1)    blocks = 1;
    scan_kernel<<<blocks, 256, 0, stream>>>(src, dst, op, E, cnt, listF, listB, cap_ws);
    gat_kernel<<<1, 256, 0, stream>>>(feat, W_f, aL_f, aR_f, b_f,
                                      W_b, aL_b, aR_b, b_b,
                                      src, dst, op, par,
                                      cnt, listF, listB, cap_ws, (float*)d_out);
}